// BoundaryCalculationModule_50354196578521
// MI455X (gfx1250) — compile-verified
//
#include <hip/hip_runtime.h>
#include <cstdint>

#define IMG_W 512
#define IMG_H 512
#define TILE_H 16
#define HALO_ROWS (TILE_H + 2)
#define NTHREADS 256

typedef float nfloat4 __attribute__((ext_vector_type(4)));  // native vec for builtins

__global__ __launch_bounds__(NTHREADS) void
boundary_edge_kernel(const float* __restrict__ in, float* __restrict__ out)
{
    // 18 rows x 512 cols x 4B = 36 KB LDS tile (incl. top/bottom halo)
    __shared__ float tile[HALO_ROWS * IMG_W];

    const int tid   = threadIdx.x;
    const int tileY = blockIdx.x;            // 0..31 (row tiles)
    const int b     = blockIdx.y;            // 0..31 (batch)
    const float* src = in + (size_t)b * (IMG_H * IMG_W);
    const int gy0 = tileY * TILE_H - 1;      // global row of LDS row 0

    // ---------- Stage 18 rows into LDS via async global->LDS (16B/lane) ----------
    // Static mapping: quad column fixed per thread, row advances by 2 per step.
    const int q4    = (tid & 127) * 4;       // first column of this thread's quad
    const int rbase = tid >> 7;              // 0 or 1
    {
        const uint32_t lds_base = (uint32_t)(uintptr_t)&tile[rbase * IMG_W + q4];
        const int gy_first = gy0 + rbase;
        #pragma unroll
        for (int i = 0; i < 9; ++i) {
            const int gy = gy_first + 2 * i;             // global input row
            if ((unsigned)gy < (unsigned)IMG_H) {
                const uint32_t lds_off = lds_base + (uint32_t)(i * 2 * IMG_W * 4);
                const uint32_t voff    = (uint32_t)((gy * IMG_W + q4) * 4);
                asm volatile("global_load_async_to_lds_b128 %0, %1, %2"
                             :: "v"(lds_off), "v"(voff), "s"(src)
                             : "memory");
            } else {
                // SAME padding: zero-fill halo rows outside the image
                nfloat4 z = {0.f, 0.f, 0.f, 0.f};
                *reinterpret_cast<nfloat4*>(&tile[(rbase + 2 * i) * IMG_W + q4]) = z;
            }
        }
    }
    asm volatile("s_wait_asynccnt 0" ::: "memory");
    __syncthreads();

    // ---------- Compute: separable directional stencil ----------
    // Thread owns a 4-wide column strip, 8 consecutive output rows.
    const int xq   = q4;
    const int ly0  = 1 + rbase * 8;          // LDS row of first output row
    const int gyo0 = tileY * TILE_H + rbase * 8;

    // Horizontal sums for one LDS row: r1=[1,2,1], r2=[1,0,-1], r3=[1,1,1]
    auto hsum = [&](int lr, nfloat4& r1, nfloat4& r2, nfloat4& r3) {
        const float* row = &tile[lr * IMG_W + xq];
        nfloat4 c = *reinterpret_cast<const nfloat4*>(row);    // ds_load_b128
        float L = (xq == 0)         ? 0.f : row[-1];           // SAME zero pad
        float R = (xq == IMG_W - 4) ? 0.f : row[4];
        float t0 = L    + c[1]; r1[0] = fmaf(2.f, c[0], t0); r2[0] = L    - c[1]; r3[0] = t0 + c[0];
        float t1 = c[0] + c[2]; r1[1] = fmaf(2.f, c[1], t1); r2[1] = c[0] - c[2]; r3[1] = t1 + c[1];
        float t2 = c[1] + c[3]; r1[2] = fmaf(2.f, c[2], t2); r2[2] = c[1] - c[3]; r3[2] = t2 + c[2];
        float t3 = c[2] + R;    r1[3] = fmaf(2.f, c[3], t3); r2[3] = c[2] - R;    r3[3] = t3 + c[3];
    };

    nfloat4 r1t, r2t, r3t, r1m, r2m, r3m, r1b, r2b, r3b;
    hsum(ly0 - 1, r1t, r2t, r3t);
    hsum(ly0,     r1m, r2m, r3m);

    #pragma unroll
    for (int k = 0; k < 8; ++k) {
        hsum(ly0 + 1 + k, r1b, r2b, r3b);

        nfloat4 o;
        #pragma unroll
        for (int j = 0; j < 4; ++j) {
            float e0 = r1t[j] - r1b[j];            // K0 (vert sobel)
            float t2 = r2t[j] + r2b[j];
            float e1 = fmaf(2.f, r2m[j], t2);      // K1 (horiz sobel)
            float u  = r3t[j] - r3b[j];            // K2 = u+v, K3 = u-v
            float v  = t2 + r2m[j];
            float s1 = fmaf(e0, e0, e1 * e1);
            float s2 = fmaf(u, u, v * v);
            float s  = fmaf(2.f, s2, s1);          // e2^2+e3^2 = 2(u^2+v^2)
            float edge = __builtin_amdgcn_sqrtf(s) * 0.125f; // /4 kern, /2
            o[j] = __builtin_amdgcn_rcpf(1.0f + __expf(-edge));
        }

        nfloat4* dst = reinterpret_cast<nfloat4*>(
            out + ((size_t)b * IMG_H + (gyo0 + k)) * IMG_W + xq);
        __builtin_nontemporal_store(o, dst);     // global_store_b128 th:NT

        r1t = r1m; r2t = r2m; r3t = r3m;
        r1m = r1b; r2m = r2b; r3m = r3b;
    }
}

extern "C" void kernel_launch(void* const* d_in, const int* in_sizes, int n_in,
                              void* d_out, int out_size, void* d_ws, size_t ws_size,
                              hipStream_t stream)
{
    (void)in_sizes; (void)n_in; (void)out_size; (void)d_ws; (void)ws_size;
    const float* pred = (const float*)d_in[0];   // (32,1,512,512) f32
    float* out = (float*)d_out;                  // (32,1,512,512) f32

    dim3 grid(IMG_H / TILE_H, 32);               // 32 row-tiles x 32 batch
    dim3 block(NTHREADS);
    boundary_edge_kernel<<<grid, block, 0, stream>>>(pred, out);
}